// NormalLoss_1382979470110
// MI455X (gfx1250) — compile-verified
//
#include <hip/hip_runtime.h>
#include <math.h>

typedef __attribute__((ext_vector_type(2))) float v2f;
typedef __attribute__((ext_vector_type(8))) float v8f;

#define NN 10            // neighborhood size
#define BATCH 4

// ---------------------------------------------------------------------------
// Fused KNN (via WMMA Gram tiles) + PCA-normal kernel.
// One wave (32 threads) handles 16 query points of one batch.
// Score ranked per candidate n: s = |p_n|^2 - 2 p_m.p_n  (row-constant |p_m|^2
// dropped -> same ordering as true squared distance, self included like ref).
// ---------------------------------------------------------------------------
__global__ __launch_bounds__(32)
void knn_normals_kernel(const float* __restrict__ pts,
                        float* __restrict__ normals, int Npts)
{
    __shared__ float tile[16 * 17];              // pitch 17: conflict-free r/w
    const int lane  = threadIdx.x;               // 0..31 (wave32)
    const int tilesPerBatch = Npts >> 4;
    const int b     = blockIdx.x / tilesPerBatch;
    const int r0    = (blockIdx.x % tilesPerBatch) << 4;
    const int mrow  = lane & 15;
    const int hi    = lane >> 4;                 // 0: K=0,1 | 1: K=2,3

    const float* batch = pts + (size_t)b * Npts * 3;

    // A (16x4, f32): lanes 0-15 carry (x,y) of row M=lane, lanes 16-31 carry
    // (z,0) of row M=lane-16. Pre-scaled by -2 so WMMA emits the score directly.
    const float* qp = batch + (size_t)(r0 + mrow) * 3;
    const float qx = qp[0], qy = qp[1], qz = qp[2];
    v2f av;
    av.x = -2.0f * (hi ? qz : qx);
    av.y = -2.0f * (hi ? 0.0f : qy);

    float best_d[NN]; int best_i[NN];
#pragma unroll
    for (int k = 0; k < NN; ++k) { best_d[k] = 3.4e38f; best_i[k] = 0; }

    for (int c0 = 0; c0 < Npts; c0 += 16) {
        // prefetch next column tile (global_prefetch_b8)
        if (c0 + 16 < Npts)
            __builtin_prefetch(batch + (size_t)(c0 + 16) * 3, 0, 1);

        // B (4x16, f32): same per-lane packing, column N = mrow of this tile.
        const float* cp = batch + (size_t)(c0 + mrow) * 3;
        const float cx = cp[0], cy = cp[1], cz = cp[2];
        const float sqc = cx * cx + cy * cy + cz * cz;
        v2f bv;
        bv.x = hi ? cz : cx;
        bv.y = hi ? 0.0f : cy;

        // Seed accumulator with |p_n|^2 (column-constant == lane-constant).
        v8f cc;
#pragma unroll
        for (int v = 0; v < 8; ++v) cc[v] = sqc;

        // score(m,n) = |p_n|^2 - 2 p_m . p_n
        cc = __builtin_amdgcn_wmma_f32_16x16x4_f32(false, av, false, bv,
                                                   (short)0, cc, false, false);

        // C layout: VGPR v holds (M = v + 8*hi, N = mrow). Transpose via LDS.
#pragma unroll
        for (int v = 0; v < 8; ++v)
            tile[(v + 8 * hi) * 17 + mrow] = cc[v];
        __syncthreads();

        // Batch-load the whole row into registers (contiguous in LDS ->
        // ds_load_2addr_b32 pairs, ONE dscnt wait) then scan registers.
        // Every lane ranks row (lane&15); upper half duplicates work so EXEC
        // stays all-1s around the WMMA.
        float rv[16];
#pragma unroll
        for (int j = 0; j < 16; ++j) rv[j] = tile[mrow * 17 + j];

#pragma unroll
        for (int j = 0; j < 16; ++j) {
            const float d = rv[j];
            if (d < best_d[NN - 1]) {              // rarely taken after warm-up
                best_d[NN - 1] = d; best_i[NN - 1] = c0 + j;
#pragma unroll
                for (int s = NN - 1; s > 0; --s) {
                    if (best_d[s] < best_d[s - 1]) {
                        const float td = best_d[s]; best_d[s] = best_d[s-1]; best_d[s-1] = td;
                        const int   ti = best_i[s]; best_i[s] = best_i[s-1]; best_i[s-1] = ti;
                    }
                }
            }
        }
        __syncthreads();
    }

    // ---- gather neighborhood, 3x3 covariance ----
    float nx[NN], ny[NN], nz[NN];
    float mx = 0.f, my = 0.f, mz = 0.f;
#pragma unroll
    for (int k = 0; k < NN; ++k) {
        const float* pp = batch + (size_t)best_i[k] * 3;
        nx[k] = pp[0]; ny[k] = pp[1]; nz[k] = pp[2];
        mx += nx[k]; my += ny[k]; mz += nz[k];
    }
    const float inv = 1.0f / NN;
    mx *= inv; my *= inv; mz *= inv;
    float cxx = 0.f, cxy = 0.f, cxz = 0.f, cyy = 0.f, cyz = 0.f, czz = 0.f;
#pragma unroll
    for (int k = 0; k < NN; ++k) {
        const float dx = nx[k] - mx, dy = ny[k] - my, dz = nz[k] - mz;
        cxx += dx * dx; cxy += dx * dy; cxz += dx * dz;
        cyy += dy * dy; cyz += dy * dz; czz += dz * dz;
    }
    cxx *= inv; cxy *= inv; cxz *= inv; cyy *= inv; cyz *= inv; czz *= inv;

    // ---- smallest eigenvalue (trigonometric method, symmetric 3x3) ----
    const float q  = (cxx + cyy + czz) * (1.0f / 3.0f);
    const float p1 = cxy * cxy + cxz * cxz + cyz * cyz;
    const float axx = cxx - q, ayy = cyy - q, azz = czz - q;
    const float p2 = axx * axx + ayy * ayy + azz * azz + 2.0f * p1;
    const float p  = sqrtf(p2 * (1.0f / 6.0f)) + 1e-30f;
    const float ip = 1.0f / p;
    const float bxx = axx * ip, byy = ayy * ip, bzz = azz * ip;
    const float bxy = cxy * ip, bxz = cxz * ip, byz = cyz * ip;
    float r = 0.5f * (bxx * (byy * bzz - byz * byz)
                    - bxy * (bxy * bzz - byz * bxz)
                    + bxz * (bxy * byz - byy * bxz));
    r = fminf(1.0f, fmaxf(-1.0f, r));
    const float phi = acosf(r) * (1.0f / 3.0f);
    const float e1  = q + 2.0f * p * cosf(phi + 2.0943951023931953f); // smallest

    // ---- eigenvector: largest cross product of rows of (C - e1*I) ----
    const float r0x = cxx - e1, r0y = cxy,      r0z = cxz;
    const float r1x = cxy,      r1y = cyy - e1, r1z = cyz;
    const float r2x = cxz,      r2y = cyz,      r2z = czz - e1;
    float v0x = r0y * r1z - r0z * r1y, v0y = r0z * r1x - r0x * r1z, v0z = r0x * r1y - r0y * r1x;
    float v1x = r0y * r2z - r0z * r2y, v1y = r0z * r2x - r0x * r2z, v1z = r0x * r2y - r0y * r2x;
    float v2x = r1y * r2z - r1z * r2y, v2y = r1z * r2x - r1x * r2z, v2z = r1x * r2y - r1y * r2x;
    const float n0 = v0x * v0x + v0y * v0y + v0z * v0z;
    const float n1 = v1x * v1x + v1y * v1y + v1z * v1z;
    const float n2 = v2x * v2x + v2y * v2y + v2z * v2z;
    float vx = v0x, vy = v0y, vz = v0z, nn = n0;
    if (n1 > nn) { vx = v1x; vy = v1y; vz = v1z; nn = n1; }
    if (n2 > nn) { vx = v2x; vy = v2y; vz = v2z; nn = n2; }
    if (nn < 1e-30f) { vx = 0.f; vy = 0.f; vz = 1.f; nn = 1.f; }
    const float rn = rsqrtf(nn);
    vx *= rn; vy *= rn; vz *= rn;

    if (lane < 16) {
        float* o = normals + ((size_t)b * Npts + r0 + mrow) * 3;
        o[0] = vx; o[1] = vy; o[2] = vz;
    }
}

// ---------------------------------------------------------------------------
// pred_g[b][n] = pred[b][idx12[b][n]]
// ---------------------------------------------------------------------------
__global__ void gather_kernel(const float* __restrict__ pred,
                              const int* __restrict__ idx,
                              float* __restrict__ out, int Npts, int total)
{
    const int t = blockIdx.x * blockDim.x + threadIdx.x;
    if (t >= total) return;
    const int b = t / Npts;
    const int j = idx[t];
    const float* s = pred + ((size_t)b * Npts + j) * 3;
    float* d = out + (size_t)t * 3;
    d[0] = s[0]; d[1] = s[1]; d[2] = s[2];
}

// ---------------------------------------------------------------------------
// loss partial sums: 1 - cos(n_pred, n_gt), CosineSimilarity eps = 1e-8
// ---------------------------------------------------------------------------
__global__ __launch_bounds__(256)
void loss_partial_kernel(const float* __restrict__ gn,
                         const float* __restrict__ pn,
                         float* __restrict__ partial, int total)
{
    __shared__ float red[256];
    const int t = blockIdx.x * blockDim.x + threadIdx.x;
    float loss = 0.f;
    if (t < total) {
        const float* a = gn + (size_t)t * 3;
        const float* b = pn + (size_t)t * 3;
        const float dot = a[0] * b[0] + a[1] * b[1] + a[2] * b[2];
        const float na = sqrtf(a[0] * a[0] + a[1] * a[1] + a[2] * a[2]);
        const float nb = sqrtf(b[0] * b[0] + b[1] * b[1] + b[2] * b[2]);
        loss = 1.0f - dot / fmaxf(na * nb, 1e-8f);
    }
    red[threadIdx.x] = loss;
    __syncthreads();
    for (int s = 128; s > 0; s >>= 1) {
        if (threadIdx.x < s) red[threadIdx.x] += red[threadIdx.x + s];
        __syncthreads();
    }
    if (threadIdx.x == 0) partial[blockIdx.x] = red[0];
}

__global__ __launch_bounds__(256)
void finalize_kernel(const float* __restrict__ partial, float* __restrict__ out,
                     int nparts, float scale)
{
    __shared__ float red[256];
    float s = 0.f;
    for (int i = threadIdx.x; i < nparts; i += 256) s += partial[i];
    red[threadIdx.x] = s;
    __syncthreads();
    for (int st = 128; st > 0; st >>= 1) {
        if (threadIdx.x < st) red[threadIdx.x] += red[threadIdx.x + st];
        __syncthreads();
    }
    if (threadIdx.x == 0) out[0] = red[0] * scale;
}

extern "C" void kernel_launch(void* const* d_in, const int* in_sizes, int n_in,
                              void* d_out, int out_size, void* d_ws, size_t ws_size,
                              hipStream_t stream)
{
    const float* gt    = (const float*)d_in[0];
    const float* pred  = (const float*)d_in[1];
    const int*   idx12 = (const int*)d_in[2];

    const int total = in_sizes[2];          // B * N
    const int Npts  = total / BATCH;        // N (8192)

    float* ws      = (float*)d_ws;
    float* pred_g  = ws;                                // total*3 floats
    float* gtn     = pred_g + (size_t)total * 3;        // total*3
    float* pn      = gtn    + (size_t)total * 3;        // total*3
    float* partial = pn     + (size_t)total * 3;        // nb floats

    const int nb = (total + 255) / 256;
    const int knnBlocks = BATCH * (Npts / 16);

    gather_kernel<<<nb, 256, 0, stream>>>(pred, idx12, pred_g, Npts, total);
    knn_normals_kernel<<<knnBlocks, 32, 0, stream>>>(gt, gtn, Npts);
    knn_normals_kernel<<<knnBlocks, 32, 0, stream>>>(pred_g, pn, Npts);
    loss_partial_kernel<<<nb, 256, 0, stream>>>(gtn, pn, partial, total);
    finalize_kernel<<<1, 256, 0, stream>>>(partial, (float*)d_out, nb, 1.0f / total);
}